// UpsampleBPCA3D_8332236554530
// MI455X (gfx1250) — compile-verified
//
#include <hip/hip_runtime.h>

typedef float v2f __attribute__((ext_vector_type(2)));
typedef float v8f __attribute__((ext_vector_type(8)));

// Reference dims (fixed): b=2, c=32, s=w=h=64, K=2.
// nvox = 2*32*64^3 = 16777216, out = (2,32,128,128,128) fp32.
//
// out_flat = bc*128^3 + (2d+k0)*128^2 + (2y+k1)*128 + (2z+k2)
//          = bc*2097152 + d*32768 + k0*16384 + y*256 + k1*128 + 2z + k2
//
// One wave handles 32 consecutive voxels (32-aligned => bc,d,y uniform,
// z = zb + {0..31}).  One V_WMMA_F32_16X16X4_F32 computes all 256 outputs:
//   A[m,0] = x[v0+m],  A[m,1] = x[v0+16+m]   (m = 0..15)
//   B[0,n] = (n<8) ? p[n] : 0,  B[1,n] = (n>=8) ? p[n-8] : 0, rows 2,3 = 0
//   C[m,n] = mean[n&7]
// => D[m,n] = x[v0 + m + 16*(n>=8)] * p[n&7] + mean[n&7]
// D layout: VGPR r, lane l -> voxel v0 + r + 8*(l>=16) + 16*((l&15)>=8), j = l&7.

__global__ __launch_bounds__(256, 4)
void upsample_pca3d_wmma(const float* __restrict__ x,
                         const float* __restrict__ mean8,
                         const float* __restrict__ pca8,
                         float* __restrict__ out,
                         int nvox) {
    const int lane        = threadIdx.x & 31;
    const int wavesPerBlk = blockDim.x >> 5;
    const int gwave       = blockIdx.x * wavesPerBlk + (threadIdx.x >> 5);
    const int nwaves      = gridDim.x * wavesPerBlk;
    const int nchunks     = nvox >> 5;

    // ---- per-lane constants ------------------------------------------------
    const int j  = lane & 7;                 // output index (k0,k1,k2)
    const int k0 = (j >> 2) & 1;
    const int k1 = (j >> 1) & 1;
    const int k2 = j & 1;
    const int laneVox = ((lane >= 16) ? 8 : 0) + (((lane & 15) >= 8) ? 16 : 0);
    const int laneOff = (k0 << 14) + (k1 << 7) + k2 + (laneVox << 1);

    const float pj = pca8[j];                // pca_components[j][0]
    const float mj = mean8[j];

    // B operand: VGPR0 = rows {0 (lanes 0-15), 2 (lanes 16-31)},
    //            VGPR1 = rows {1, 3}; rows 2,3 zeroed.
    v2f B;
    B.x = (lane < 8) ? pj : 0.0f;                   // B[0,n] = p[n] for n<8
    B.y = (lane >= 8 && lane < 16) ? pj : 0.0f;     // B[1,n] = p[n-8] for n>=8
    v8f C;
    #pragma unroll
    for (int r = 0; r < 8; ++r) C[r] = mj;          // C[m,n] = mean[n&7]

    for (int ch = gwave; ch < nchunks; ch += nwaves) {
        const int v0 = ch << 5;

        // One coalesced 128B read of 32 voxels; input is read exactly once,
        // so use a non-temporal load to avoid polluting L2 (output needs it).
        const float v = __builtin_nontemporal_load(&x[v0 + lane]);

        // Prefetch this wave's next chunk (speculative, silently dropped OOB).
        __builtin_prefetch(&x[(size_t)v0 + lane + ((size_t)nwaves << 5)], 0, 0);

        // a1[l] = v[l+16]  (cross-lane gather; wave32 => index bits [6:2])
        const float vsw = __int_as_float(
            __builtin_amdgcn_ds_bpermute(((lane + 16) & 31) << 2,
                                         __float_as_int(v)));
        v2f A;
        A.x = (lane < 16) ? v   : 0.0f;   // K=0 (lanes 0-15); K=2 lanes zeroed
        A.y = (lane < 16) ? vsw : 0.0f;   // K=1 (lanes 0-15); K=3 lanes zeroed

        // EXEC is all-1s here (no divergence in this loop) as WMMA requires.
        v8f D = __builtin_amdgcn_wmma_f32_16x16x4_f32(
            /*neg_a=*/false, A, /*neg_b=*/false, B,
            /*c_mod=*/(short)0, C, /*reuse_a=*/false, /*reuse_b=*/false);

        // ---- addressing (uniform per chunk except low z bits) --------------
        const int  bc = v0 >> 18;          // (b*32+c)
        const int  d  = (v0 >> 12) & 63;
        const int  y  = (v0 >> 6) & 63;
        const int  zb = v0 & 63;           // 0 or 32
        const long base = (long)bc * 2097152 + ((long)d << 15) + (y << 8)
                        + laneOff + (zb << 1);

        // 8 per-lane b32 stores; across the wave they exactly cover
        // 4 output rows x 256B each => full-line writeback from L2.
        #pragma unroll
        for (int r = 0; r < 8; ++r) {
            out[base + (r << 1)] = D[r];
        }
    }
}

extern "C" void kernel_launch(void* const* d_in, const int* in_sizes, int n_in,
                              void* d_out, int out_size, void* d_ws, size_t ws_size,
                              hipStream_t stream) {
    const float* x     = (const float*)d_in[0];   // (2,32,64,64,64) fp32
    const float* mean8 = (const float*)d_in[1];   // (8,) fp32
    const float* pca8  = (const float*)d_in[2];   // (8,1) fp32
    // d_in[3] = output_shape (int32[5]) — dims are static per the reference.
    float* out = (float*)d_out;

    const int nvox = in_sizes[0];                 // 16777216
    dim3 blk(256);
    dim3 grd(4096);                               // 32768 waves, grid-stride
    hipLaunchKernelGGL(upsample_pca3d_wmma, grd, blk, 0, stream,
                       x, mean8, pca8, out, nvox);
}